// GINAutoregressive_53979148976513
// MI455X (gfx1250) — compile-verified
//
#include <hip/hip_runtime.h>
#include <hip/hip_bf16.h>
#include <math.h>

// ---------------------------------------------------------------------------
// GIN autoregressive TSP log-likelihood, collapsed form.
// Complete-graph GIN (eps=0): pre-MLP feature = global sum => node-uniform.
// Layer-0 input has 3 distinct rows (label 0/1/2); label counts depend only
// on step t => 192 distinct rows total. Masked log-softmax contribution per
// step is logit0 - (logit0 + log(V - t)).
// ---------------------------------------------------------------------------

typedef __attribute__((ext_vector_type(2))) float v2f;
typedef __attribute__((ext_vector_type(8))) float v8f;
typedef int v4i __attribute__((vector_size(16)));

#if defined(__AMDGCN__) && __has_builtin(__builtin_amdgcn_wmma_f32_16x16x4_f32)
#define USE_WMMA_F32 1
#else
#define USE_WMMA_F32 0
#endif

#if defined(__AMDGCN__) && __has_builtin(__builtin_amdgcn_global_load_async_to_lds_b128)
#define HAVE_ASYNC_LDS 1
typedef __attribute__((address_space(1))) v4i as1_v4i;   // global
typedef __attribute__((address_space(3))) v4i as3_v4i;   // LDS
#else
#define HAVE_ASYNC_LDS 0
#endif

__device__ __forceinline__ void wait_async0() {
#if defined(__AMDGCN__)
#if __has_builtin(__builtin_amdgcn_s_wait_asynccnt)
  __builtin_amdgcn_s_wait_asynccnt(0);
#else
  asm volatile("s_wait_asynccnt 0" ::: "memory");
#endif
#endif
}

#define HH   128
#define VV   64
#define LDSP 132   // LDS row stride for activations (132 % 64 banks = 4)
#define KCH  16    // K-chunk staged into LDS per step

// C = relu( src(48x128) @ W(128x128) + bias ).
// Weights staged global->LDS in KCH-row chunks via async-LDS DMA.
// Called uniformly by all 256 threads (EXEC all 1s for WMMA; barriers inside).
__device__ __forceinline__ void gemm48(const float (*src)[LDSP], float (*dst)[LDSP],
                                       float (*wb)[HH],
                                       const float* __restrict__ W,
                                       const float* __restrict__ bias,
                                       int lane, int wave, int tid) {
#if USE_WMMA_F32
  const int nb  = wave << 4;          // this wave's 16-wide N column
  const int lid = lane & 15;
  const int khi = (lane >> 4) << 1;   // lanes 0-15 -> K+0/K+1, lanes 16-31 -> K+2/K+3
  v8f acc0 = {}, acc1 = {}, acc2 = {};
  for (int kc = 0; kc < HH / KCH; ++kc) {
    // ---- stage W rows [kc*KCH, kc*KCH+KCH) into wb ----
    const float* gsrc = W + kc * (KCH * HH);
#if HAVE_ASYNC_LDS
    {
      // KCH*HH floats = 512 x b128 ; 2 per thread. ASYNCcnt tracks completion.
      char* lbase = (char*)&wb[0][0];
      const int j0 = tid * 2;
      __builtin_amdgcn_global_load_async_to_lds_b128(
          (as1_v4i*)(gsrc + j0 * 4),       (as3_v4i*)(lbase + j0 * 16),       0, 0);
      __builtin_amdgcn_global_load_async_to_lds_b128(
          (as1_v4i*)(gsrc + (j0 + 1) * 4), (as3_v4i*)(lbase + (j0 + 1) * 16), 0, 0);
      wait_async0();   // this wave's LDS writes landed; barrier covers the rest
    }
#else
    for (int j = tid; j < KCH * HH; j += 256) wb[j >> 7][j & (HH - 1)] = gsrc[j];
#endif
    __syncthreads();
    // ---- KCH/4 WMMA K-steps from the staged chunk ----
    for (int k0 = 0; k0 < KCH; k0 += 4) {
      const int kl = k0 + khi;            // K row within chunk
      const int kr = kc * KCH + kl;       // K row global (for A fragments)
      v2f bf;                             // B fragment: 4x16 from LDS-staged W
      bf.x = wb[kl + 0][nb + lid];
      bf.y = wb[kl + 1][nb + lid];
      v2f a0, a1, a2;                     // A fragments: 16x4, M = lane%16
      a0.x = src[lid][kr];       a0.y = src[lid][kr + 1];
      a1.x = src[16 + lid][kr];  a1.y = src[16 + lid][kr + 1];
      a2.x = src[32 + lid][kr];  a2.y = src[32 + lid][kr + 1];
      acc0 = __builtin_amdgcn_wmma_f32_16x16x4_f32(false, a0, false, bf, (short)0, acc0, false, false);
      acc1 = __builtin_amdgcn_wmma_f32_16x16x4_f32(false, a1, false, bf, (short)0, acc1, false, false);
      acc2 = __builtin_amdgcn_wmma_f32_16x16x4_f32(false, a2, false, bf, (short)0, acc2, false, false);
    }
    __syncthreads();   // protect wb before next chunk overwrites it
  }
  // C/D layout: VGPR j -> M=j (lanes 0-15) / M=8+j (lanes 16-31), N = lane%16
  const int   col = nb + lid;
  const float bv  = bias[col];
  const int   ro  = (lane >> 4) << 3;     // 0 or 8
  for (int j = 0; j < 8; ++j) {
    dst[j + ro][col]      = fmaxf(acc0[j] + bv, 0.f);
    dst[16 + j + ro][col] = fmaxf(acc1[j] + bv, 0.f);
    dst[32 + j + ro][col] = fmaxf(acc2[j] + bv, 0.f);
  }
#else
  (void)wb;
  for (int i = tid; i < 48 * HH; i += 256) {
    const int r = i >> 7, c = i & (HH - 1);
    float s = bias[c];
    for (int k = 0; k < HH; ++k) s += src[r][k] * W[k * HH + c];
    dst[r][c] = fmaxf(s, 0.f);
  }
#endif
}

// Each block handles 16 step-contexts: rows 3*tt+label, tt in [0,16).
__global__ __launch_bounds__(256) void gin_ctx_mlp_kernel(
    const float* __restrict__ emb,  const float* __restrict__ W1,
    const float* __restrict__ b1,   const float* __restrict__ W2,
    const float* __restrict__ b2,   const float* __restrict__ w_out,
    const float* __restrict__ b_out, float* __restrict__ logits_ws) {
  __shared__ float xs[48][LDSP];
  __shared__ float hs[48][LDSP];
  __shared__ __align__(16) float wb[KCH][HH];   // async-staged weight K-chunk
  const int tid   = threadIdx.x;
  const int lane  = tid & 31;
  const int wave  = tid >> 5;
  const int tbase = blockIdx.x * 16;

  // layer-0 input: 3 label rows per context
  for (int i = tid; i < 48 * HH; i += 256) {
    const int row = i >> 7, col = i & (HH - 1);
    xs[row][col] = emb[(row % 3) * HH + col];
  }
  __syncthreads();

  for (int l = 0; l < 5; ++l) {
    // warm L2 for this layer's weights (global_prefetch_b8)
    {
      const char* w1p = (const char*)(W1 + l * HH * HH);
      const char* w2p = (const char*)(W2 + l * HH * HH);
      for (int off = tid * 64; off < HH * HH * 4; off += 256 * 64) {
        __builtin_prefetch(w1p + off, 0, 1);
        __builtin_prefetch(w2p + off, 0, 1);
      }
    }
    // pre-MLP aggregate: S = n0*x0 + n1*x1 + n2*x2 ; x_i <- x_i + (S - x_i)
    for (int i = tid; i < 16 * HH; i += 256) {
      const int tt = i >> 7, col = i & (HH - 1);
      const int t  = tbase + tt;
      const float n0 = (t == 0) ? (float)VV : (float)(VV - t);
      const float n1 = (t == 0) ? 0.f : (float)(t - 1);
      const float n2 = (t == 0) ? 0.f : 1.f;
      const float x0 = xs[3 * tt + 0][col];
      const float x1 = xs[3 * tt + 1][col];
      const float x2 = xs[3 * tt + 2][col];
      const float S  = n0 * x0 + n1 * x1 + n2 * x2;
      xs[3 * tt + 0][col] = x0 + (S - x0);
      xs[3 * tt + 1][col] = x1 + (S - x1);
      xs[3 * tt + 2][col] = x2 + (S - x2);
    }
    __syncthreads();
    gemm48(xs, hs, wb, W1 + l * HH * HH, b1 + l * HH, lane, wave, tid); // relu(xW1+b1)
    __syncthreads();
    gemm48(hs, xs, wb, W2 + l * HH * HH, b2 + l * HH, lane, wave, tid); // relu(hW2+b2)
    __syncthreads();
  }

  // logits: 48 rows dotted with w_out; 8 waves x 6 rows
  const float bo = b_out[0];
  for (int r = wave * 6; r < wave * 6 + 6; ++r) {
    float p = 0.f;
    for (int c = lane; c < HH; c += 32) p += xs[r][c] * w_out[c];
    for (int off = 16; off > 0; off >>= 1) p += __shfl_down(p, off, 32);
    if (lane == 0) logits_ws[(tbase + r / 3) * 3 + (r % 3)] = p + bo;
  }
}

// Masked log-softmax accumulation; result is batch-independent.
__global__ void gin_lp_kernel(const float* __restrict__ logits_ws,
                              float* __restrict__ out) {
  __shared__ float contrib[VV];
  const int t = threadIdx.x;
  if (t < VV) {
    const float l0 = logits_ws[t * 3 + 0];
    float c;
    if (t == 0) {
      c = l0 - (l0 + logf((float)VV));            // all 64 nodes available
    } else {
      const float l1 = logits_ws[t * 3 + 1] - 1e9f;
      const float l2 = logits_ws[t * 3 + 2] - 1e9f;
      const float m  = fmaxf(l0, fmaxf(l1, l2));  // == l0
      const float s  = (float)(VV - t) * expf(l0 - m)
                     + (float)(t - 1)  * expf(l1 - m)   // underflows to 0
                     +                   expf(l2 - m);  // underflows to 0
      c = l0 - (m + logf(s));
    }
    contrib[t] = c;
  }
  __syncthreads();
  if (t == 0) {
    float lp = 0.f;
    for (int i = 0; i < VV; ++i) lp += contrib[i];
    contrib[0] = lp;
  }
  __syncthreads();
  if (t < 32) out[t] = contrib[0];
}

extern "C" void kernel_launch(void* const* d_in, const int* in_sizes, int n_in,
                              void* d_out, int out_size, void* d_ws, size_t ws_size,
                              hipStream_t stream) {
  (void)in_sizes; (void)n_in; (void)out_size; (void)ws_size;
  // dict order: samples, emb, W1, b1, W2, b2, w_out, b_out
  const float* emb  = (const float*)d_in[1];
  const float* W1   = (const float*)d_in[2];
  const float* b1   = (const float*)d_in[3];
  const float* W2   = (const float*)d_in[4];
  const float* b2   = (const float*)d_in[5];
  const float* wout = (const float*)d_in[6];
  const float* bout = (const float*)d_in[7];
  float* logits_ws  = (float*)d_ws;   // 64 * 3 floats

  gin_ctx_mlp_kernel<<<4, 256, 0, stream>>>(emb, W1, b1, W2, b2, wout, bout, logits_ws);
  gin_lp_kernel<<<1, 64, 0, stream>>>(logits_ws, (float*)d_out);
}